// HierarchicalAttentionNetworkCRF_82686710382736
// MI455X (gfx1250) — compile-verified
//
#include <hip/hip_runtime.h>

// Problem constants (from reference setup_inputs)
#define BDOCS 512
#define SSENT 128
#define DFEAT 1200
#define TTAG  4
#define NEG_INF_F (-10000.0f)

typedef float v2f __attribute__((ext_vector_type(2)));
typedef float v8f __attribute__((ext_vector_type(8)));

// ---------------------------------------------------------------------------
// Kernel 1: feats[r, t] = sum_k s_c[r, k] * fc_W[t, k] + fc_b[t]
// r = b*128 + s  (65536 rows), t in [0,4) -- N padded to 16 for WMMA.
// One wave per 16-row tile; 300 x V_WMMA_F32_16X16X4_F32 per tile.
// HBM-bound: streams s_c (314.6 MB) exactly once with NT hint.
// LDS holds fc_W plus a 5th all-zero row so padded B columns (n>=4) read
// broadcast zeros straight into the WMMA (no per-iteration mask VALU).
// ---------------------------------------------------------------------------
__global__ __launch_bounds__(256)
void feats_gemm_wmma(const float* __restrict__ s_c,
                     const float* __restrict__ fc_W,
                     const float* __restrict__ fc_b,
                     float* __restrict__ feats) {
    __shared__ float Wlds[(TTAG + 1) * DFEAT];     // 24 KB (row 4 = zeros)
    for (int i = threadIdx.x; i < TTAG * DFEAT; i += 256) Wlds[i] = fc_W[i];
    for (int i = threadIdx.x; i < DFEAT; i += 256) Wlds[TTAG * DFEAT + i] = 0.0f;
    __syncthreads();

    const int lane    = threadIdx.x & 31;
    const int wave    = threadIdx.x >> 5;
    const int tile    = blockIdx.x * 8 + wave;     // 512 blocks * 8 waves = 4096 tiles
    const int rowBase = tile * 16;
    const int half    = lane >> 4;                 // 0 or 1 (K sub-pair select)
    const int n       = lane & 15;                 // A-row within tile == B/C column
    const int wrow    = (n < TTAG) ? n : TTAG;     // padded cols -> zero row

    // A fragment: lane holds row (rowBase+n), cols {base, base+1}, base = k0 + 2*half
    const float* arow = s_c + (size_t)(rowBase + n) * DFEAT + 2 * half;
    // B fragment: lane holds fc_W[n][base..base+1] (or zeros for n>=4)
    const float* brow = &Wlds[wrow * DFEAT + 2 * half];

    v8f c = {};
    #pragma unroll 8
    for (int k0 = 0; k0 < DFEAT; k0 += 4) {
        v2f a = __builtin_nontemporal_load((const v2f*)(arow + k0)); // b64, TH=NT
        v2f b = *(const v2f*)(brow + k0);                            // ds_load_b64
        // (neg_a, A, neg_b, B, c_mod, C, reuse_a, reuse_b)
        c = __builtin_amdgcn_wmma_f32_16x16x4_f32(false, a, false, b,
                                                  (short)0, c, false, false);
    }

    // C/D layout: VGPR r -> M = r + 8*half, N = lane&15
    if (n < TTAG) {
        const float bias = fc_b[n];
        const int m0 = rowBase + 8 * half;
        #pragma unroll
        for (int r = 0; r < 8; ++r) {
            feats[(size_t)(m0 + r) * TTAG + n] = c[r] + bias;
        }
    }
}

// ---------------------------------------------------------------------------
// Kernel 2: Viterbi forward scan + backtrace. One lane per batch element.
// Backpointers: 4 tags x 2 bits packed into one byte per step, in LDS.
// ---------------------------------------------------------------------------
__global__ __launch_bounds__(256)
void viterbi_kernel(const float* __restrict__ feats,
                    const float* __restrict__ trans,       // [4][4] (next, prev)
                    float* __restrict__ path_partials,     // per-block partial sums
                    int* __restrict__ best_paths,          // [512*128]
                    float* __restrict__ tag_probs) {       // [512*128*2]
    __shared__ unsigned char bp[SSENT * 256];              // 32 KB, [s][tid]
    __shared__ float red[256];

    const int tid = threadIdx.x;
    const int b   = blockIdx.x * 256 + tid;                // batch element

    float tr[16];
    #pragma unroll
    for (int i = 0; i < 16; ++i) tr[i] = trans[i];

    float v[4] = {NEG_INF_F, NEG_INF_F, 0.0f, NEG_INF_F}; // START_TAG = 2

    for (int s = 0; s < SSENT; ++s) {
        const float* f = feats + ((size_t)b * SSENT + s) * TTAG;
        float f0 = f[0], f1 = f[1], f2 = f[2], f3 = f[3];

        float nv[4]; int nbp[4];
        #pragma unroll
        for (int nx = 0; nx < 4; ++nx) {
            float best = v[0] + tr[nx * 4 + 0]; int bi = 0;
            #pragma unroll
            for (int p = 1; p < 4; ++p) {
                float sc = v[p] + tr[nx * 4 + p];
                if (sc > best) { best = sc; bi = p; }   // first-max tiebreak = argmax
            }
            nv[nx] = best; nbp[nx] = bi;
        }
        nv[0] += f0; nv[1] += f1; nv[2] += f2; nv[3] += f3;
        v[0] = nv[0]; v[1] = nv[1]; v[2] = nv[2]; v[3] = nv[3];

        bp[s * 256 + tid] = (unsigned char)(nbp[0] | (nbp[1] << 2) |
                                            (nbp[2] << 4) | (nbp[3] << 6));
        tag_probs[((size_t)b * SSENT + s) * 2 + 0] = nv[0];
        tag_probs[((size_t)b * SSENT + s) * 2 + 1] = nv[1];
    }

    // terminal: v + transitions[-1] (row 3)
    float best = v[0] + tr[3 * 4 + 0]; int tag = 0;
    #pragma unroll
    for (int p = 1; p < 4; ++p) {
        float sc = v[p] + tr[3 * 4 + p];
        if (sc > best) { best = sc; tag = p; }
    }

    // backtrace: path[S-1] = best_last; path[s-1] = bp[s][path[s]]
    best_paths[(size_t)b * SSENT + (SSENT - 1)] = tag;
    for (int s = SSENT - 1; s >= 1; --s) {
        unsigned int code = bp[s * 256 + tid];
        tag = (int)((code >> (2 * tag)) & 3u);
        best_paths[(size_t)b * SSENT + s - 1] = tag;
    }

    // deterministic block reduction of best terminal scores
    red[tid] = best;
    __syncthreads();
    #pragma unroll
    for (int off = 128; off > 0; off >>= 1) {
        if (tid < off) red[tid] += red[tid + off];
        __syncthreads();
    }
    if (tid == 0) path_partials[blockIdx.x] = red[0];
}

// ---------------------------------------------------------------------------
// Kernel 3: sum the (2) block partials into d_out[0]. Deterministic.
// ---------------------------------------------------------------------------
__global__ void finalize_score(const float* __restrict__ partials, int nparts,
                               float* __restrict__ out0) {
    if (threadIdx.x == 0 && blockIdx.x == 0) {
        float s = 0.0f;
        for (int i = 0; i < nparts; ++i) s += partials[i];
        *out0 = s;
    }
}

extern "C" void kernel_launch(void* const* d_in, const int* in_sizes, int n_in,
                              void* d_out, int out_size, void* d_ws, size_t ws_size,
                              hipStream_t stream) {
    (void)in_sizes; (void)n_in; (void)out_size; (void)ws_size;
    const float* s_c   = (const float*)d_in[0];   // [512,128,1200]
    const float* fc_W  = (const float*)d_in[1];   // [4,1200]
    const float* fc_b  = (const float*)d_in[2];   // [4]
    const float* trans = (const float*)d_in[3];   // [4,4]

    float* out = (float*)d_out;
    // Output layout (concatenated in return order):
    float* path_score = out;                                    // [1]
    int*   best_paths = (int*)(out + 1);                        // [512*128] int32
    float* tag_probs  = out + 1 + (size_t)BDOCS * SSENT;        // [512*128*2]

    // Workspace: feats [65536*4] floats (1 MB), then partial sums
    float* feats    = (float*)d_ws;
    float* partials = feats + (size_t)BDOCS * SSENT * TTAG;

    feats_gemm_wmma<<<512, 256, 0, stream>>>(s_c, fc_W, fc_b, feats);
    viterbi_kernel<<<(BDOCS + 255) / 256, 256, 0, stream>>>(feats, trans, partials,
                                                            best_paths, tag_probs);
    finalize_score<<<1, 32, 0, stream>>>(partials, (BDOCS + 255) / 256, path_score);
}